// MultiBoxLoss_6949257085128
// MI455X (gfx1250) — compile-verified
//
#include <hip/hip_runtime.h>
#include <stdint.h>
#include <math.h>

// Problem dims (fixed by the reference's setup_inputs)
#define BB 64
#define AA 8732
#define CC 81
#define GG 32
#define TA 512    // kernel A block (16 waves; per-thread 32 u64 maxima -> keep pressure low)
#define TC 1024   // kernel C block
#define NCHUNK ((AA + 3) / 4)   // 2183 x b128 transfers per batch row (A*4 bytes /16)

typedef float v2f __attribute__((ext_vector_type(2)));
typedef float v8f __attribute__((ext_vector_type(8)));

__device__ __forceinline__ float warp_red_sum_f(float v) {
  for (int o = 16; o > 0; o >>= 1) v += __shfl_xor(v, o, 32);
  return v;
}
__device__ __forceinline__ float warp_red_max_f(float v) {
  for (int o = 16; o > 0; o >>= 1) v = fmaxf(v, __shfl_xor(v, o, 32));
  return v;
}
__device__ __forceinline__ int warp_red_sum_i(int v) {
  for (int o = 16; o > 0; o >>= 1) v += __shfl_xor(v, o, 32);
  return v;
}

// ---------------------------------------------------------------------------
// Kernel A: per-batch matching + loc smooth-L1.  grid = BB blocks of TA.
// ---------------------------------------------------------------------------
__global__ __launch_bounds__(TA)
void kA_match(const float* __restrict__ loc_pred, const float* __restrict__ anchors,
              const float* __restrict__ gt_boxes, const int* __restrict__ gt_labels,
              int* __restrict__ labels_ws, float* __restrict__ loc_sum_ws,
              int* __restrict__ num_pos_ws) {
  const int b = blockIdx.x;
  const int tid = threadIdx.x;
  const int lane = tid & 31, wid = tid >> 5;

  __shared__ float g_cx[GG], g_cy[GG], g_w[GG], g_h[GG];
  __shared__ float g_x0[GG], g_y0[GG], g_x1[GG], g_y1[GG], g_area[GG];
  __shared__ int   g_lab[GG];
  __shared__ unsigned long long best_pa[GG];   // per-gt best anchor: (iou_bits<<32)|(~a)
  __shared__ int   best_anchor[GG];
  __shared__ unsigned int packA[AA];           // per-anchor: best_gt | over<<31
  __shared__ float redf[TA / 32];
  __shared__ int   redi[TA / 32];

  if (tid < GG) {
    float cx = gt_boxes[(b * GG + tid) * 4 + 0];
    float cy = gt_boxes[(b * GG + tid) * 4 + 1];
    float w  = gt_boxes[(b * GG + tid) * 4 + 2];
    float h  = gt_boxes[(b * GG + tid) * 4 + 3];
    g_cx[tid] = cx; g_cy[tid] = cy; g_w[tid] = w; g_h[tid] = h;
    float x0 = cx - w * 0.5f, y0 = cy - h * 0.5f;
    float x1 = cx + w * 0.5f, y1 = cy + h * 0.5f;
    g_x0[tid] = x0; g_y0[tid] = y0; g_x1[tid] = x1; g_y1[tid] = y1;
    g_area[tid] = (x1 - x0) * (y1 - y0);        // area from xyxy (match reference)
    g_lab[tid] = gt_labels[b * GG + tid];
    best_pa[tid] = 0xFFFFFFFFull;               // iou=0, anchor 0
  }
  __syncthreads();

  unsigned long long mybest[GG];
  #pragma unroll
  for (int g = 0; g < GG; ++g) mybest[g] = 0ull;

  for (int a = tid; a < AA; a += TA) {
    float acx = anchors[a * 4 + 0], acy = anchors[a * 4 + 1];
    float aw  = anchors[a * 4 + 2], ah  = anchors[a * 4 + 3];
    float ax0 = acx - aw * 0.5f, ay0 = acy - ah * 0.5f;
    float ax1 = acx + aw * 0.5f, ay1 = acy + ah * 0.5f;
    float areaA = (ax1 - ax0) * (ay1 - ay0);
    float bov = -1.0f; int bg = 0;
    #pragma unroll 4
    for (int g = 0; g < GG; ++g) {
      float ltx = fmaxf(g_x0[g], ax0), lty = fmaxf(g_y0[g], ay0);
      float rbx = fminf(g_x1[g], ax1), rby = fminf(g_y1[g], ay1);
      float w = fmaxf(rbx - ltx, 0.f), h = fmaxf(rby - lty, 0.f);
      float inter = w * h;
      float iou = inter / (g_area[g] + areaA - inter);
      if (iou > bov) { bov = iou; bg = g; }      // first-max == jnp.argmax(0)
      unsigned long long cand =
          ((unsigned long long)__float_as_uint(iou) << 32) |
          (unsigned long long)(0xFFFFFFFFu - (unsigned)a);  // tie -> smallest a
      if (cand > mybest[g]) mybest[g] = cand;
    }
    packA[a] = (unsigned)bg | (bov > 0.5f ? 0x80000000u : 0u);
  }
  // per-gt best anchor: wave reduce then one LDS u64 atomic-max per wave
  #pragma unroll 4
  for (int g = 0; g < GG; ++g) {
    unsigned long long v = mybest[g];
    for (int o = 16; o > 0; o >>= 1) {
      unsigned long long u = __shfl_xor(v, o, 32);
      if (u > v) v = u;
    }
    if (lane == 0) atomicMax(&best_pa[g], v);
  }
  __syncthreads();
  if (tid < GG)
    best_anchor[tid] = (int)(0xFFFFFFFFu - (unsigned)(best_pa[tid] & 0xFFFFFFFFull));
  __syncthreads();

  // pass 2: targets, labels, loc smooth-L1
  float locsum = 0.f; int pcnt = 0;
  for (int a = tid; a < AA; a += TA) {
    float acx = anchors[a * 4 + 0], acy = anchors[a * 4 + 1];
    float aw  = anchors[a * 4 + 2], ah  = anchors[a * 4 + 3];
    unsigned p = packA[a];
    int bg = (int)(p & 0x7FFFFFFFu);
    int lab = 0;
    float e0 = 0.f, e1 = 0.f, e2 = 0.f, e3 = 0.f;
    if (p & 0x80000000u) {
      lab = g_lab[bg];
      e0 = (g_cx[bg] - acx) / aw; e1 = (g_cy[bg] - acy) / ah;
      e2 = logf(g_w[bg]) - logf(aw); e3 = logf(g_h[bg]) - logf(ah);
    }
    #pragma unroll 4
    for (int g = 0; g < GG; ++g) {   // ascending: last gt wins (scatter .set)
      if (best_anchor[g] == a) {
        lab = g_lab[g];
        e0 = (g_cx[g] - acx) / aw; e1 = (g_cy[g] - acy) / ah;
        e2 = logf(g_w[g]) - logf(aw); e3 = logf(g_h[g]) - logf(ah);
      }
    }
    labels_ws[(size_t)b * AA + a] = lab;
    if (lab > 0) {
      ++pcnt;
      const float* lp = loc_pred + (size_t)(b * AA + a) * 4;
      float t[4] = {e0, e1, e2, e3};
      #pragma unroll
      for (int j = 0; j < 4; ++j) {
        float d = lp[j] - t[j];
        float ad = fabsf(d);
        locsum += (ad < 1.f) ? 0.5f * d * d : (ad - 0.5f);
      }
    }
  }
  locsum = warp_red_sum_f(locsum);
  int pc = warp_red_sum_i(pcnt);
  if (lane == 0) { redf[wid] = locsum; redi[wid] = pc; }
  __syncthreads();
  if (tid == 0) {
    float s = 0.f; int c = 0;
    for (int i = 0; i < TA / 32; ++i) { s += redf[i]; c += redi[i]; }
    loc_sum_ws[b] = s; num_pos_ws[b] = c;
  }
}

// ---------------------------------------------------------------------------
// Kernel B: one wave per anchor; lse / conf[0] / conf[label] in ONE pass over
// the 181MB conf tensor (bandwidth floor ~7.8us @ 23.3 TB/s).
// ---------------------------------------------------------------------------
__global__ __launch_bounds__(256)
void kB_conf(const float* __restrict__ conf, const int* __restrict__ labels_ws,
             float* __restrict__ cls_l_ws, float* __restrict__ posce_partial) {
  const int lane = threadIdx.x & 31, wid = threadIdx.x >> 5;
  const int ga = blockIdx.x * 8 + wid;           // exact: (BB*AA)%8==0
  __shared__ float wsum[8];
  const float* cp = conf + (size_t)ga * CC;
  float x0 = cp[lane];
  float x1 = cp[lane + 32];
  float x2 = (lane < CC - 64) ? cp[lane + 64] : -__builtin_inff();
  float m = warp_red_max_f(fmaxf(fmaxf(x0, x1), x2));
  float s = expf(x0 - m) + expf(x1 - m) + ((lane < CC - 64) ? expf(x2 - m) : 0.f);
  s = warp_red_sum_f(s);
  float lse = m + logf(s);
  float c0 = __shfl(x0, 0, 32);
  int lab = labels_ws[ga];
  float t = 0.f;
  if (lane == lab) t = x0;
  else if (lane + 32 == lab) t = x1;
  else if (lane < CC - 64 && lane + 64 == lab) t = x2;
  t = warp_red_sum_f(t);                         // exactly one lane contributes
  bool pos = lab > 0;
  if (lane == 0) {
    cls_l_ws[ga] = pos ? 0.f : (lse - c0);       // (lse - conf[idx01])*(1-pos)
    wsum[wid] = pos ? (lse - t) : 0.f;           // positive CE contribution
  }
  __syncthreads();
  if (threadIdx.x == 0) {
    float acc = 0.f;
    for (int i = 0; i < 8; ++i) acc += wsum[i];
    posce_partial[blockIdx.x] = acc;
  }
}

// ---------------------------------------------------------------------------
// Kernel C: hard-negative mining via exact radix-select in LDS.
// cls_l row (35KB) staged with CDNA5 async global->LDS (ASYNCcnt path).
// ---------------------------------------------------------------------------
__global__ __launch_bounds__(TC)
void kC_mine(const float* __restrict__ cls_l_ws, const int* __restrict__ labels_ws,
             const int* __restrict__ num_pos_ws,
             float* __restrict__ neg_ce_ws, float* __restrict__ sampled_ws) {
  const int b = blockIdx.x;
  const int tid = threadIdx.x;
  const int lane = tid & 31, wid = tid >> 5;

  __shared__ float sc[AA];                 // 34928 B (easy inside 320KB WGP LDS)
  __shared__ unsigned int hist[256];
  __shared__ unsigned int sel[2];
  __shared__ unsigned int tiecnt;
  __shared__ unsigned int ties[256];       // (index<<1)|posflag
  __shared__ float redf[TC / 32];
  __shared__ int   redi[TC / 32];

  const float* src = cls_l_ws + (size_t)b * AA;

  // --- CDNA5 async global -> LDS copy (b128 per lane), then wait ASYNCcnt ---
  unsigned int ldsbase = (unsigned int)(uintptr_t)(&sc[0]);  // LDS offset = addr[31:0]
  for (int c = tid; c < NCHUNK; c += TC) {
    unsigned int lo = ldsbase + (unsigned int)c * 16u;
    unsigned long long gaddr = (unsigned long long)(uintptr_t)(src + c * 4);
    asm volatile("global_load_async_to_lds_b128 %0, %1, off"
                 :: "v"(lo), "v"(gaddr) : "memory");
  }
  asm volatile("s_wait_asynccnt 0" ::: "memory");
  __syncthreads();

  int npos = num_pos_ws[b];
  float nnf = fminf(3.0f * (float)npos, (float)(AA - 1));    // num_neg (integer-valued)
  int k = (int)nnf;
  if (k <= 0) {
    if (tid == 0) { neg_ce_ws[b] = 0.f; sampled_ws[b] = (float)npos; }
    return;
  }

  // 4-pass radix select for k-th largest (all cls_l >= 0 => bit order == numeric)
  unsigned int prefix = 0, mask = 0, kk = (unsigned)k;
  for (int shift = 24; shift >= 0; shift -= 8) {
    __syncthreads();
    if (tid < 256) hist[tid] = 0u;
    __syncthreads();
    for (int a = tid; a < AA; a += TC) {
      unsigned u = __float_as_uint(sc[a]);
      if ((u & mask) == prefix) atomicAdd(&hist[(u >> shift) & 0xFFu], 1u);
    }
    __syncthreads();
    if (tid == 0) {
      unsigned cum = 0, bin = 0, nk = kk;
      for (int i = 255; i >= 0; --i) {
        cum += hist[i];
        if (cum >= kk) { bin = (unsigned)i; nk = kk - (cum - hist[i]); break; }
      }
      sel[0] = bin; sel[1] = nk;
    }
    __syncthreads();
    prefix |= sel[0] << shift;
    mask   |= 0xFFu << shift;
    kk = sel[1];
  }
  const unsigned thrbits = prefix;
  const float thr = __uint_as_float(prefix);

  if (tid == 0) tiecnt = 0u;
  __syncthreads();

  float sumgt = 0.f; int cntgt = 0;
  for (int a = tid; a < AA; a += TC) {
    float v = sc[a];
    unsigned u = __float_as_uint(v);
    if (u > thrbits) { sumgt += v; ++cntgt; }
    else if (u == thrbits) {
      unsigned slot = atomicAdd(&tiecnt, 1u);
      if (slot < 256u) {
        int posf = (labels_ws[(size_t)b * AA + a] > 0) ? 1 : 0;
        ties[slot] = ((unsigned)a << 1) | (unsigned)posf;
      }
    }
  }
  sumgt = warp_red_sum_f(sumgt);
  cntgt = warp_red_sum_i(cntgt);
  if (lane == 0) { redf[wid] = sumgt; redi[wid] = cntgt; }
  __syncthreads();
  if (tid == 0) {
    float sg = 0.f; int cg = 0;
    for (int i = 0; i < TC / 32; ++i) { sg += redf[i]; cg += redi[i]; }
    int r = k - cg; if (r < 0) r = 0;                 // stable ties to take
    unsigned tc = tiecnt; if (tc > 256u) tc = 256u;
    for (unsigned i = 1; i < tc; ++i) {               // sort ties by index (stable rank)
      unsigned key = ties[i]; int j = (int)i - 1;
      while (j >= 0 && ties[j] > key) { ties[j + 1] = ties[j]; --j; }
      ties[j + 1] = key;
    }
    int take = (r < (int)tc) ? r : (int)tc;
    int extra = 0;                                    // non-positive ties selected
    for (int i = 0; i < take; ++i) if (!(ties[i] & 1u)) ++extra;
    neg_ce_ws[b]  = sg + thr * (float)extra;          // negative CE == cls_l values
    sampled_ws[b] = (float)npos + (float)cg + (float)extra;
  }
}

// ---------------------------------------------------------------------------
// Kernel D: deterministic finalize.  The 69856-partial reduction runs on the
// matrix pipe: V_WMMA_F32_16X16X4_F32 with A==ones gives D[m,n] = sum_k B[k,n]
// + C[m,n]; one WMMA folds 64 f32 partials into a persistent accumulator
// (sum is invariant to B's internal (k,n) lane mapping; D is uniform over m).
// Row M=0 of C/D = lanes 0..15, VGPR0 (documented layout) -> exact extract.
// ---------------------------------------------------------------------------
__global__ __launch_bounds__(256)
void kD_final(const float* __restrict__ loc_sum_ws, const int* __restrict__ num_pos_ws,
              const float* __restrict__ neg_ce_ws, const float* __restrict__ sampled_ws,
              const float* __restrict__ posce_partial, int nPart,
              float* __restrict__ out) {
  __shared__ float wtot[8];
  const int lane = threadIdx.x & 31, wid = threadIdx.x >> 5;
  const int stripe = (nPart + 7) / 8;          // 8732 per wave (8*8732 == nPart)
  const int start  = wid * stripe;
  const int endi   = (start + stripe < nPart) ? (start + stripe) : nPart;
  const int iters  = (stripe + 63) / 64;       // uniform trip count: EXEC all-1s at WMMA

  v8f c = {};
  const v2f ones = {1.f, 1.f};
  for (int it = 0; it < iters; ++it) {
    int i0 = start + it * 64 + lane;
    int i1 = i0 + 32;
    float x0 = (i0 < endi) ? posce_partial[i0] : 0.f;   // per-lane select, no branch
    float x1 = (i1 < endi) ? posce_partial[i1] : 0.f;
    v2f bv = {x0, x1};
    c = __builtin_amdgcn_wmma_f32_16x16x4_f32(false, ones, false, bv,
                                              (short)0, c, false, false);
  }
  float v = (lane < 16) ? c[0] : 0.f;          // row M=0: lanes 0..15, VGPR0
  v = warp_red_sum_f(v);
  if (lane == 0) wtot[wid] = v;
  __syncthreads();
  if (threadIdx.x == 0) {
    float posce = 0.f;
    for (int i = 0; i < 8; ++i) posce += wtot[i];
    float loc = 0.f, neg = 0.f, samp = 0.f; int N = 0;
    for (int b = 0; b < BB; ++b) {
      loc += loc_sum_ws[b]; neg += neg_ce_ws[b];
      samp += sampled_ws[b]; N += num_pos_ws[b];
    }
    // non-sampled rows contribute lse(zeros) - 0 = ln(C) each (faithful to reference)
    float cls = posce + neg + ((float)(BB * AA) - samp) * logf((float)CC);
    out[0] = (loc + cls) / (float)N;
  }
}

// ---------------------------------------------------------------------------
extern "C" void kernel_launch(void* const* d_in, const int* in_sizes, int n_in,
                              void* d_out, int out_size, void* d_ws, size_t ws_size,
                              hipStream_t stream) {
  (void)in_sizes; (void)n_in; (void)out_size; (void)ws_size;
  const float* loc_pred  = (const float*)d_in[0];
  const float* conf_pred = (const float*)d_in[1];
  const float* anchors   = (const float*)d_in[2];
  const float* gt_boxes  = (const float*)d_in[3];
  const int*   gt_labels = (const int*)d_in[4];
  float* out = (float*)d_out;

  char* ws = (char*)d_ws;
  size_t o = 0;
  int*   labels_ws   = (int*)(ws + o);   o += (size_t)BB * AA * sizeof(int);
  float* cls_l_ws    = (float*)(ws + o); o += (size_t)BB * AA * sizeof(float);
  float* loc_sum_ws  = (float*)(ws + o); o += (size_t)BB * sizeof(float);
  int*   num_pos_ws  = (int*)(ws + o);   o += (size_t)BB * sizeof(int);
  float* neg_ce_ws   = (float*)(ws + o); o += (size_t)BB * sizeof(float);
  float* sampled_ws  = (float*)(ws + o); o += (size_t)BB * sizeof(float);
  const int nB = (BB * AA) / 8;          // 69856 blocks, one wave per anchor
  float* posce_partial = (float*)(ws + o); o += (size_t)nB * sizeof(float);

  kA_match<<<BB, TA, 0, stream>>>(loc_pred, anchors, gt_boxes, gt_labels,
                                  labels_ws, loc_sum_ws, num_pos_ws);
  kB_conf<<<nB, 256, 0, stream>>>(conf_pred, labels_ws, cls_l_ws, posce_partial);
  kC_mine<<<BB, TC, 0, stream>>>(cls_l_ws, labels_ws, num_pos_ws,
                                 neg_ce_ws, sampled_ws);
  kD_final<<<1, 256, 0, stream>>>(loc_sum_ws, num_pos_ws, neg_ce_ws, sampled_ws,
                                  posce_partial, nB, out);
}